// SGC_24524263260254
// MI455X (gfx1250) — compile-verified
//
#include <hip/hip_runtime.h>
#include <math.h>

typedef float v2f __attribute__((ext_vector_type(2)));
typedef float v8f __attribute__((ext_vector_type(8)));

#define NFEAT 128
#define NCLS  40
#define NCLSP 48      // padded to 3 x 16 WMMA tiles
#define LDST  52      // LDS row stride (floats); 52 avoids bank conflicts for 16-lane row reads

// ---------------- degree / normalization ----------------

__global__ __launch_bounds__(256) void k_deg_init(float* deg, int N) {
    int i = blockIdx.x * 256 + threadIdx.x;
    if (i < N) deg[i] = 1.0f;               // self-loop contribution
}

__global__ __launch_bounds__(256) void k_deg_edges(const int* __restrict__ dst,
                                                   float* __restrict__ deg, int E) {
    int i = blockIdx.x * 256 + threadIdx.x;
    if (i < E) atomicAdd(&deg[dst[i]], 1.0f);
}

__global__ __launch_bounds__(256) void k_rsqrt(float* deg, int N) {
    int i = blockIdx.x * 256 + threadIdx.x;
    if (i < N) deg[i] = rsqrtf(deg[i]);     // deg >= 1 always (self loops)
}

// ---------------- pad W (128x40 -> 128x48) and b ----------------

__global__ __launch_bounds__(256) void k_prep_w(const float* __restrict__ W,
                                                const float* __restrict__ b,
                                                float* __restrict__ Wp,
                                                float* __restrict__ bp) {
    int i = blockIdx.x * 256 + threadIdx.x;
    if (i < NFEAT * NCLSP) {
        int k = i / NCLSP, c = i % NCLSP;
        Wp[i] = (c < NCLS) ? W[k * NCLS + c] : 0.0f;
    }
    if (i < NCLSP) bp[i] = (i < NCLS) ? b[i] : 0.0f;
}

// ---------------- propagation: Y = dis^2 * X (self loop) ----------------

__global__ __launch_bounds__(256) void k_self_init(const float* __restrict__ X,
                                                   const float* __restrict__ dis,
                                                   float* __restrict__ Y, int N) {
    long long idx = (long long)blockIdx.x * 256 + threadIdx.x;   // over N*32 float4s
    int n = (int)(idx >> 5), c = (int)(idx & 31);
    if (n < N) {
        float w = dis[n] * dis[n];
        float4 v = ((const float4*)(X + (size_t)n * NFEAT))[c];
        float4 o; o.x = w * v.x; o.y = w * v.y; o.z = w * v.z; o.w = w * v.w;
        ((float4*)(Y + (size_t)n * NFEAT))[c] = o;
    }
}

// ---------------- propagation: edge scatter, one wave per edge ----------------

__global__ __launch_bounds__(256) void k_edge_prop(const int* __restrict__ src,
                                                   const int* __restrict__ dst,
                                                   const float* __restrict__ dis,
                                                   const float* __restrict__ X,
                                                   float* __restrict__ Y, int E) {
    int wid    = (blockIdx.x * 256 + threadIdx.x) >> 5;
    int lane   = threadIdx.x & 31;
    int stride = (gridDim.x * 256) >> 5;
    for (int e = wid; e < E; e += stride) {
        int nx = e + stride;
        if (nx < E)   // prefetch next gathered row (global_prefetch_b8); L2-resident working set
            __builtin_prefetch(X + (size_t)src[nx] * NFEAT + lane * 4, 0, 0);
        int s = src[e], d = dst[e];
        float w = dis[s] * dis[d];
        float4 v = ((const float4*)(X + (size_t)s * NFEAT))[lane];
        float* yd = Y + (size_t)d * NFEAT + lane * 4;
        atomicAdd(yd + 0, w * v.x);
        atomicAdd(yd + 1, w * v.y);
        atomicAdd(yd + 2, w * v.z);
        atomicAdd(yd + 3, w * v.w);
    }
}

// ---------------- WMMA GEMM (fp32, exact) + bias + log_softmax ----------------
// Per wave: 16 rows (M-tile) x 48 cols (3 N-tiles), K=128 in steps of 4.
// A 16x4 f32: lane&15 = M, element K = 2*(lane>>4) + vgpr
// B 4x16 f32: lane&15 = N, element K = 2*(lane>>4) + vgpr
// C/D 16x16 f32: vgpr r -> M = r + 8*(lane>>4), N = lane&15

__global__ __launch_bounds__(256) void k_gemm_logsoftmax(const float* __restrict__ X,
                                                         const float* __restrict__ Wp,
                                                         const float* __restrict__ bp,
                                                         float* __restrict__ out, int M) {
    __shared__ float stage[8 * 16 * LDST];
    int wave = threadIdx.x >> 5;
    int lane = threadIdx.x & 31;
    int lh   = lane >> 4;       // lane half: 0 or 1
    int lid  = lane & 15;
    int mbase = (blockIdx.x * 8 + wave) * 16;
    bool active = (mbase < M);  // wave-uniform -> EXEC all-1s inside branch (WMMA requirement)

    if (active) {
        int arow = mbase + lid; if (arow > M - 1) arow = M - 1;   // clamp; masked at store
        const float* xr = X + (size_t)arow * NFEAT;
        const int kofs = 2 * lh;
        v8f acc0 = {}, acc1 = {}, acc2 = {};
        #pragma unroll 4
        for (int k0 = 0; k0 < NFEAT; k0 += 4) {
            v2f a;
            a.x = xr[k0 + kofs + 0];
            a.y = xr[k0 + kofs + 1];
            const float* w0 = Wp + (size_t)(k0 + kofs) * NCLSP + lid;
            const float* w1 = w0 + NCLSP;
            v2f b0, b1, b2;
            b0.x = w0[0];  b0.y = w1[0];
            b1.x = w0[16]; b1.y = w1[16];
            b2.x = w0[32]; b2.y = w1[32];
            acc0 = __builtin_amdgcn_wmma_f32_16x16x4_f32(false, a, false, b0, (short)0, acc0, false, false);
            acc1 = __builtin_amdgcn_wmma_f32_16x16x4_f32(false, a, false, b1, (short)0, acc1, false, false);
            acc2 = __builtin_amdgcn_wmma_f32_16x16x4_f32(false, a, false, b2, (short)0, acc2, false, false);
        }
        #pragma unroll
        for (int r = 0; r < 8; ++r) {
            float* srow = &stage[(wave * 16 + r + 8 * lh) * LDST];
            srow[ 0 + lid] = acc0[r];
            srow[16 + lid] = acc1[r];
            srow[32 + lid] = acc2[r];
        }
    }
    __syncthreads();
    if (active && lh == 0) {                 // lanes 0..15: one output row each
        int m = mbase + lid;
        if (m < M) {
            const float* srow = &stage[(wave * 16 + lid) * LDST];
            float mx = -INFINITY;
            for (int c = 0; c < NCLS; ++c) mx = fmaxf(mx, srow[c] + bp[c]);
            float s = 0.0f;
            for (int c = 0; c < NCLS; ++c) s += expf(srow[c] + bp[c] - mx);
            float lse = mx + logf(s);
            float* orow = out + (size_t)m * NCLS;
            for (int c = 0; c < NCLS; ++c) orow[c] = srow[c] + bp[c] - lse;
        }
    }
}

// ---------------- host launch ----------------

extern "C" void kernel_launch(void* const* d_in, const int* in_sizes, int n_in,
                              void* d_out, int out_size, void* d_ws, size_t ws_size,
                              hipStream_t stream) {
    const float* x    = (const float*)d_in[0];
    const float* W    = (const float*)d_in[1];
    const float* b    = (const float*)d_in[2];
    const int*   edge = (const int*)d_in[3];
    // d_in[4] is K (==2); propagation depth is fixed by the reference problem.

    int N = in_sizes[0] / NFEAT;
    int E = in_sizes[3] / 2;
    const int* src = edge;
    const int* dst = edge + E;

    // workspace carve-out
    char* ws = (char*)d_ws;
    size_t off = 0;
    auto carve = [&](size_t bytes) -> void* {
        void* p = ws + off;
        off = (off + bytes + 255) & ~(size_t)255;
        return p;
    };
    float* dis  = (float*)carve((size_t)N * 4);
    float* bufA = (float*)carve((size_t)N * NFEAT * 4);
    float* bufB = (float*)carve((size_t)N * NFEAT * 4);
    float* Wp   = (float*)carve((size_t)NFEAT * NCLSP * 4);
    float* bp   = (float*)carve((size_t)NCLSP * 4);
    (void)ws_size;

    int nbN = (N + 255) / 256;
    int nbE = (E + 255) / 256;
    k_deg_init <<<nbN, 256, 0, stream>>>(dis, N);
    k_deg_edges<<<nbE, 256, 0, stream>>>(dst, dis, E);
    k_rsqrt    <<<nbN, 256, 0, stream>>>(dis, N);
    k_prep_w   <<<(NFEAT * NCLSP + 255) / 256, 256, 0, stream>>>(W, b, Wp, bp);

    long long nq = (long long)N * 32;                 // float4s in the feature matrix
    int nbQ = (int)((nq + 255) / 256);
    int waves = (E + 3) / 4;                          // ~4 edges per wave
    int nbEdge = (waves + 7) / 8;                     // 8 waves per 256-thread block

    // hop 1: x -> bufA
    k_self_init<<<nbQ, 256, 0, stream>>>(x, dis, bufA, N);
    k_edge_prop<<<nbEdge, 256, 0, stream>>>(src, dst, dis, x, bufA, E);
    // hop 2: bufA -> bufB
    k_self_init<<<nbQ, 256, 0, stream>>>(bufA, dis, bufB, N);
    k_edge_prop<<<nbEdge, 256, 0, stream>>>(src, dst, dis, bufA, bufB, E);

    // head: WMMA GEMM + bias + log_softmax
    int nbG = (N + 127) / 128;                        // 8 waves x 16 rows per block
    k_gemm_logsoftmax<<<nbG, 256, 0, stream>>>(bufB, Wp, bp, (float*)d_out, N);
}